// IMUPreintegrator_43181601194817
// MI455X (gfx1250) — compile-verified
//
#include <hip/hip_runtime.h>
#include <math.h>

typedef float v2f __attribute__((ext_vector_type(2)));
typedef float v8f __attribute__((ext_vector_type(8)));
typedef int   v4i __attribute__((ext_vector_type(4)));
typedef int   v8i __attribute__((ext_vector_type(8)));

namespace {
constexpr int kB = 16;
constexpr int kF = 16384;
constexpr int kSeg = 512;
constexpr int kNseg = kF / kSeg;              // 32 segments per batch
constexpr float kG = 9.81007f;

constexpr int OFF_VEL = kB * kF * 4;          // rot is first
constexpr int OFF_POS = OFF_VEL + kB * kF * 3;
constexpr int OFF_COV = OFF_POS + kB * kF * 3;

__device__ __forceinline__ void qmul(const float a[4], const float b[4], float o[4]) {
  // xyzw convention, matches reference _quat_mul
  o[0] = a[3]*b[0] + a[0]*b[3] + a[1]*b[2] - a[2]*b[1];
  o[1] = a[3]*b[1] - a[0]*b[2] + a[1]*b[3] + a[2]*b[0];
  o[2] = a[3]*b[2] + a[0]*b[1] - a[1]*b[0] + a[2]*b[3];
  o[3] = a[3]*b[3] - a[0]*b[0] - a[1]*b[1] - a[2]*b[2];
}

__device__ __forceinline__ void q2m(const float q[4], float R[3][3]) {
  float x = q[0], y = q[1], z = q[2], w = q[3];
  R[0][0] = 1.f - 2.f*(y*y + z*z); R[0][1] = 2.f*(x*y - z*w);       R[0][2] = 2.f*(x*z + y*w);
  R[1][0] = 2.f*(x*y + z*w);       R[1][1] = 1.f - 2.f*(x*x + z*z); R[1][2] = 2.f*(y*z - x*w);
  R[2][0] = 2.f*(x*z - y*w);       R[2][1] = 2.f*(y*z + x*w);       R[2][2] = 1.f - 2.f*(x*x + y*y);
}

// Swizzled 16x16 tile addressing: element (r,c) lives at (r&7)*32 + (r>=8?16:0) + c.
// This makes the C/D-layout register v of lane L correspond exactly to index v*32+L.
__device__ __forceinline__ int tidx(int r, int c) {
  return (r & 7) * 32 + ((r >> 3) << 4) + c;
}

__device__ __forceinline__ v8f wmma4(v2f a, v2f b, v8f c) {
  // V_WMMA_F32_16X16X4_F32 : D = A(16x4) x B(4x16) + C
  return __builtin_amdgcn_wmma_f32_16x16x4_f32(false, a, false, b, (short)0, c, false, false);
}

// B-operand chunk kk (rows 4kk..4kk+3 of a 16x16 tile):
// VGPR0: lanes0-15 row K=0, lanes16-31 row K=2; VGPR1: rows K=1 / K=3. Lane%16 = column.
__device__ __forceinline__ v2f ldB(const float* t, int kk, int hi, int n) {
  int r0 = 4*kk + 2*hi;
  v2f b; b.x = t[tidx(r0, n)]; b.y = t[tidx(r0 + 1, n)]; return b;
}
// A-operand chunk kk (cols 4kk..4kk+3): lane%16 = row m; VGPR0 holds col 4kk+2*hi, VGPR1 col +1.
__device__ __forceinline__ v2f ldA(const float* t, int kk, int hi, int mm) {
  int c0 = 4*kk + 2*hi;
  v2f a; a.x = t[tidx(mm, c0)]; a.y = t[tidx(mm, c0 + 1)]; return a;
}

// Tensor Data Mover: 1-row contiguous copy of nelem f32 from global -> LDS.
// D# groups built per CDNA5 ISA (group0: count/lds/global/type=2,
// group1: data_size=4B, tensor_dim0=nelem, tensor_dim1=1, tile_dim0=nelem, tile_dim1=1).
__device__ __forceinline__ void tdm_load_1d(unsigned lds_off_bytes, const void* gptr,
                                            unsigned nelem) {
  unsigned long long ga = (unsigned long long)gptr;
  v4i g0;
  g0[0] = 1;                                               // count=1 (valid user D#)
  g0[1] = (int)lds_off_bytes;                              // lds_addr (bytes)
  g0[2] = (int)(unsigned)(ga & 0xFFFFFFFFu);               // global_addr[31:0]
  g0[3] = (int)(((unsigned)(ga >> 32) & 0x01FFFFFFu) | 0x80000000u); // [56:32] | type=2
  v8i g1;
  g1[0] = (int)(2u << 16);                                 // data_size = 4 bytes
  g1[1] = (int)((nelem & 0xFFFFu) << 16);                  // tensor_dim0[15:0]
  g1[2] = (int)(((nelem >> 16) & 0xFFFFu) | (1u << 16));   // tensor_dim0[31:16] | tensor_dim1=1
  g1[3] = (int)((nelem & 0xFFFFu) << 16);                  // tensor_dim1 hi=0 | tile_dim0
  g1[4] = 1;                                               // tile_dim1=1, tile_dim2=0
  g1[5] = (int)nelem;                                      // tensor_dim0_stride[31:0]
  g1[6] = 0;
  g1[7] = 0;
  v4i gz = {0, 0, 0, 0};                                   // unused dims (groups 2/3)
  asm volatile("tensor_load_to_lds %0, %1, %2, %2"
               :: "s"(g0), "s"(g1), "s"(gz)
               : "memory");
}

// block-wide inclusive float scan (256 threads = 8 waves), sh must hold >= 9 floats
__device__ __forceinline__ float block_scan(float x, int lane, int wid, float* sh, float* tot) {
  float lx = x;
  #pragma unroll
  for (int off = 1; off < 32; off <<= 1) {
    float o = __shfl_up(lx, off, 32);
    if (lane >= off) lx += o;
  }
  if (lane == 31) sh[wid] = lx;
  __syncthreads();
  if (threadIdx.x == 0) {
    float p = 0.f;
    for (int w = 0; w < 8; ++w) { float t = sh[w]; sh[w] = p; p += t; }
    sh[8] = p;
  }
  __syncthreads();
  float r = lx + sh[wid];
  *tot = sh[8];
  __syncthreads();
  return r;
}
} // namespace

// ---------------------------------------------------------------------------
// Kernel 1: quaternion scan + cumsums; writes rot/vel/pos and incre_r (ws)
// ---------------------------------------------------------------------------
__global__ __launch_bounds__(256) void imu_scan_kernel(
    const float* __restrict__ dt, const float* __restrict__ gyro,
    const float* __restrict__ acc, const float* __restrict__ init_pos,
    const float* __restrict__ init_rot, const float* __restrict__ init_vel,
    float* __restrict__ rot_out, float* __restrict__ vel_out,
    float* __restrict__ pos_out, float* __restrict__ qws) {
  __shared__ float shq[36];   // 8 wave quats + block total (xyzw)
  __shared__ float shf[9];    // float scan scratch
  const int b = blockIdx.x;
  const int tid = threadIdx.x;
  const int lane = tid & 31;
  const int wid = tid >> 5;

  float qi[4] = {init_rot[4*b], init_rot[4*b+1], init_rot[4*b+2], init_rot[4*b+3]};
  float Ri[3][3]; q2m(qi, Ri);
  const float p0x = init_pos[3*b], p0y = init_pos[3*b+1], p0z = init_pos[3*b+2];
  const float v0x = init_vel[3*b], v0y = init_vel[3*b+1], v0z = init_vel[3*b+2];

  float cq[4] = {0.f, 0.f, 0.f, 1.f};           // carry quaternion (incre_r so far)
  float cvx = 0, cvy = 0, cvz = 0, cpx = 0, cpy = 0, cpz = 0, cT = 0;

  for (int tile = 0; tile < kF / 256; ++tile) {
    const int k = tile * 256 + tid;
    const int idx = b * kF + k;
    if (tile + 1 < kF / 256) {
      // hide next tile's load latency behind this tile's scan chain
      __builtin_prefetch(dt + idx + 256, 0, 1);
      __builtin_prefetch(gyro + 3 * (idx + 256), 0, 1);
      __builtin_prefetch(acc + 3 * (idx + 256), 0, 1);
    }
    const float d = dt[idx];
    const float gx = gyro[3*idx], gy = gyro[3*idx+1], gz = gyro[3*idx+2];
    const float ax = acc[3*idx],  ay = acc[3*idx+1],  az = acc[3*idx+2];

    // dr = so3_exp(gyro*dt)
    float px = gx * d, py = gy * d, pz = gz * d;
    float t2 = px*px + py*py + pz*pz;
    float tt = sqrtf(t2 + 1e-24f);
    float sc = (t2 > 1e-6f) ? (sinf(0.5f * tt) / tt) : (0.5f - t2 * (1.f / 48.f));
    float lq[4] = {px * sc, py * sc, pz * sc, cosf(0.5f * tt)};

    // ---- block quaternion scan ----
    #pragma unroll
    for (int off = 1; off < 32; off <<= 1) {
      float o0 = __shfl_up(lq[0], off, 32), o1 = __shfl_up(lq[1], off, 32);
      float o2 = __shfl_up(lq[2], off, 32), o3 = __shfl_up(lq[3], off, 32);
      if (lane >= off) {
        float pr[4] = {o0, o1, o2, o3}, r[4];
        qmul(pr, lq, r);
        lq[0] = r[0]; lq[1] = r[1]; lq[2] = r[2]; lq[3] = r[3];
      }
    }
    float e[4];
    e[0] = __shfl_up(lq[0], 1, 32); e[1] = __shfl_up(lq[1], 1, 32);
    e[2] = __shfl_up(lq[2], 1, 32); e[3] = __shfl_up(lq[3], 1, 32);
    if (lane == 31) {
      shq[4*wid] = lq[0]; shq[4*wid+1] = lq[1]; shq[4*wid+2] = lq[2]; shq[4*wid+3] = lq[3];
    }
    __syncthreads();
    if (tid == 0) {
      float p[4] = {0.f, 0.f, 0.f, 1.f};
      for (int w = 0; w < 8; ++w) {
        float t0[4] = {shq[4*w], shq[4*w+1], shq[4*w+2], shq[4*w+3]};
        shq[4*w] = p[0]; shq[4*w+1] = p[1]; shq[4*w+2] = p[2]; shq[4*w+3] = p[3];
        float np[4]; qmul(p, t0, np);
        p[0] = np[0]; p[1] = np[1]; p[2] = np[2]; p[3] = np[3];
      }
      shq[32] = p[0]; shq[33] = p[1]; shq[34] = p[2]; shq[35] = p[3];
    }
    __syncthreads();
    float wp[4]   = {shq[4*wid], shq[4*wid+1], shq[4*wid+2], shq[4*wid+3]};
    float btot[4] = {shq[32], shq[33], shq[34], shq[35]};
    float bincl[4]; qmul(wp, lq, bincl);
    float bexcl[4];
    if (lane == 0) { bexcl[0] = wp[0]; bexcl[1] = wp[1]; bexcl[2] = wp[2]; bexcl[3] = wp[3]; }
    else           { qmul(wp, e, bexcl); }
    float gq[4]; qmul(cq, bincl, gq);      // global inclusive incre_r at k
    float rp[4]; qmul(cq, bexcl, rp);      // r_prev_q at k
    float ncq[4]; qmul(cq, btot, ncq);     // carry for next tile
    __syncthreads();

    float gn = rsqrtf(gq[0]*gq[0] + gq[1]*gq[1] + gq[2]*gq[2] + gq[3]*gq[3] + 1e-24f);
    float nq[4] = {gq[0]*gn, gq[1]*gn, gq[2]*gn, gq[3]*gn};
    float rn = rsqrtf(rp[0]*rp[0] + rp[1]*rp[1] + rp[2]*rp[2] + rp[3]*rp[3] + 1e-24f);
    rp[0] *= rn; rp[1] *= rn; rp[2] *= rn; rp[3] *= rn;
    float cn = rsqrtf(ncq[0]*ncq[0] + ncq[1]*ncq[1] + ncq[2]*ncq[2] + ncq[3]*ncq[3] + 1e-24f);
    cq[0] = ncq[0]*cn; cq[1] = ncq[1]*cn; cq[2] = ncq[2]*cn; cq[3] = ncq[3]*cn;

    // Ra = Rp @ acc
    float Rp[3][3]; q2m(rp, Rp);
    float Rax = Rp[0][0]*ax + Rp[0][1]*ay + Rp[0][2]*az;
    float Ray = Rp[1][0]*ax + Rp[1][1]*ay + Rp[1][2]*az;
    float Raz = Rp[2][0]*ax + Rp[2][1]*ay + Rp[2][2]*az;

    // incre_v = cumsum(Ra*dt)
    float avx = Rax * d, avy = Ray * d, avz = Raz * d;
    float tvx, tvy, tvz;
    float ivx = cvx + block_scan(avx, lane, wid, shf, &tvx);
    float ivy = cvy + block_scan(avy, lane, wid, shf, &tvy);
    float ivz = cvz + block_scan(avz, lane, wid, shf, &tvz);
    float pvx = ivx - avx, pvy = ivy - avy, pvz = ivz - avz;  // v_prev (exclusive)

    // incre_p = cumsum(v_prev*dt + 0.5*Ra*dt^2)
    float apx = pvx * d + 0.5f * Rax * d * d;
    float apy = pvy * d + 0.5f * Ray * d * d;
    float apz = pvz * d + 0.5f * Raz * d * d;
    float tpx, tpy, tpz;
    float ipx = cpx + block_scan(apx, lane, wid, shf, &tpx);
    float ipy = cpy + block_scan(apy, lane, wid, shf, &tpy);
    float ipz = cpz + block_scan(apz, lane, wid, shf, &tpz);

    float tD;
    float Dt = cT + block_scan(d, lane, wid, shf, &tD);

    // outputs
    float ro[4]; qmul(qi, nq, ro);
    rot_out[4*idx] = ro[0]; rot_out[4*idx+1] = ro[1]; rot_out[4*idx+2] = ro[2]; rot_out[4*idx+3] = ro[3];
    qws[4*idx] = nq[0]; qws[4*idx+1] = nq[1]; qws[4*idx+2] = nq[2]; qws[4*idx+3] = nq[3];

    float Dvx = Ri[0][0]*ivx + Ri[0][1]*ivy + Ri[0][2]*ivz;
    float Dvy = Ri[1][0]*ivx + Ri[1][1]*ivy + Ri[1][2]*ivz;
    float Dvz = Ri[2][0]*ivx + Ri[2][1]*ivy + Ri[2][2]*ivz;
    vel_out[3*idx]   = v0x + Dvx;
    vel_out[3*idx+1] = v0y + Dvy;
    vel_out[3*idx+2] = v0z + Dvz + kG * Dt;

    float Dpx = Ri[0][0]*ipx + Ri[0][1]*ipy + Ri[0][2]*ipz;
    float Dpy = Ri[1][0]*ipx + Ri[1][1]*ipy + Ri[1][2]*ipz;
    float Dpz = Ri[2][0]*ipx + Ri[2][1]*ipy + Ri[2][2]*ipz;
    pos_out[3*idx]   = p0x + Dpx + v0x * Dt;
    pos_out[3*idx+1] = p0y + Dpy + v0y * Dt;
    pos_out[3*idx+2] = p0z + Dpz + v0z * Dt + 0.5f * kG * Dt * Dt;

    cvx += tvx; cvy += tvy; cvz += tvz;
    cpx += tpx; cpy += tpy; cpz += tpz;
    cT  += tD;
  }
}

// ---------------------------------------------------------------------------
// Kernel 2: per-segment affine-map (A,Q) composition with f32 WMMA.
// Inputs for the whole 512-step segment are staged into LDS by the TDM.
// ---------------------------------------------------------------------------
__global__ __launch_bounds__(32) void imu_seg_kernel(
    const float* __restrict__ dt, const float* __restrict__ gyro,
    const float* __restrict__ acc, const float* __restrict__ gyro_cov,
    const float* __restrict__ acc_cov, const float* __restrict__ qws,
    float* __restrict__ segres) {
  // single allocation so LDS byte offsets are known (slab starts at offset 0)
  __shared__ float slab[6404];
  float* sDt = slab;          // floats [   0..511 ], LDS bytes     0
  float* sGy = slab + 512;    // floats [ 512..2047], LDS bytes  2048
  float* sAc = slab + 2048;   // floats [2048..3583], LDS bytes  8192
  float* sQw = slab + 3584;   // floats [3584..5635], LDS bytes 14336 (513 quats)
  float* tM  = slab + 5636;
  float* tS  = slab + 5892;
  float* tT  = slab + 6148;

  const int b = blockIdx.x / kNseg;
  const int s = blockIdx.x % kNseg;
  const int lane = threadIdx.x;
  const int m = lane & 15;
  const int hi = lane >> 4;

  const int start = b * kF + s * kSeg;             // global step index of segment start
  const int qstart = (start == 0) ? 0 : start - 1; // first quat staged
  const int qshift = (start == 0) ? -1 : 0;        // sQw index of step j's r_prev = j + qshift

  // ---- async TDM staging of the whole segment's inputs into LDS ----
  tdm_load_1d(0u,     dt + start,        512u);
  tdm_load_1d(2048u,  gyro + 3 * start, 1536u);
  tdm_load_1d(8192u,  acc + 3 * start,  1536u);
  tdm_load_1d(14336u, qws + 4 * qstart, 2052u);

  const float gc0 = gyro_cov[0], gc1 = gyro_cov[1], gc2 = gyro_cov[2];
  const float ac0 = acc_cov[0],  ac1 = acc_cov[1],  ac2 = acc_cov[2];

  v8f M, S = {};
  #pragma unroll
  for (int v = 0; v < 8; ++v) M[v] = ((v + 8*hi) == m) ? 1.f : 0.f;  // identity (C-layout)

  __builtin_amdgcn_s_wait_tensorcnt(0);  // TDM writes to LDS complete
  __syncthreads();

  for (int j = 0; j < kSeg; ++j) {
    const int k = start + j;

    // stage current M, S into LDS (B-operand source)
    #pragma unroll
    for (int v = 0; v < 8; ++v) { tM[v*32 + lane] = M[v]; tS[v*32 + lane] = S[v]; }
    __syncthreads();

    // ---- per-step scalar matrices (wave-uniform; compiler scalarizes to SALU) ----
    const float d = sDt[j];
    const float gx = sGy[3*j], gy = sGy[3*j+1], gz = sGy[3*j+2];
    const float ax = sAc[3*j], ay = sAc[3*j+1], az = sAc[3*j+2];
    float px = gx * d, py = gy * d, pz = gz * d;
    float t2 = px*px + py*py + pz*pz;
    float tt = sqrtf(t2 + 1e-24f);
    float sc = (t2 > 1e-6f) ? (sinf(0.5f * tt) / tt) : (0.5f - t2 * (1.f / 48.f));
    float q[4] = {px * sc, py * sc, pz * sc, cosf(0.5f * tt)};
    float Rdr[3][3]; q2m(q, Rdr);

    float rpq[4];
    if (k == 0) { rpq[0] = 0.f; rpq[1] = 0.f; rpq[2] = 0.f; rpq[3] = 1.f; }
    else {
      const int qo = 4 * (j + qshift);
      rpq[0] = sQw[qo]; rpq[1] = sQw[qo+1]; rpq[2] = sQw[qo+2]; rpq[3] = sQw[qo+3];
    }
    float Rp[3][3]; q2m(rpq, Rp);

    float RpK[3][3];
    #pragma unroll
    for (int i = 0; i < 3; ++i) {
      RpK[i][0] = Rp[i][1]*az - Rp[i][2]*ay;
      RpK[i][1] = Rp[i][2]*ax - Rp[i][0]*az;
      RpK[i][2] = Rp[i][0]*ay - Rp[i][1]*ax;
    }
    float c1 = (t2 > 1e-6f) ? (1.f - cosf(tt)) / (t2 + 1e-24f) : 0.5f - t2 * (1.f / 24.f);
    float c2 = (t2 > 1e-6f) ? (tt - sinf(tt)) / (t2 * tt + 1e-24f) : (1.f / 6.f) - t2 * (1.f / 120.f);
    float K[3][3] = {{0.f, -pz, py}, {pz, 0.f, -px}, {-py, px, 0.f}};
    float K2[3][3], Jr[3][3];
    #pragma unroll
    for (int i = 0; i < 3; ++i)
      #pragma unroll
      for (int jj = 0; jj < 3; ++jj)
        K2[i][jj] = K[i][0]*K[0][jj] + K[i][1]*K[1][jj] + K[i][2]*K[2][jj];
    #pragma unroll
    for (int i = 0; i < 3; ++i)
      #pragma unroll
      for (int jj = 0; jj < 3; ++jj)
        Jr[i][jj] = ((i == jj) ? 1.f : 0.f) - c1 * K[i][jj] + c2 * K2[i][jj];

    float Sg[3][3], Sa[3][3];
    #pragma unroll
    for (int i = 0; i < 3; ++i)
      #pragma unroll
      for (int jj = 0; jj < 3; ++jj) {
        Sg[i][jj] = Jr[i][0]*gc0*Jr[jj][0] + Jr[i][1]*gc1*Jr[jj][1] + Jr[i][2]*gc2*Jr[jj][2];
        Sa[i][jj] = Rp[i][0]*ac0*Rp[jj][0] + Rp[i][1]*ac1*Rp[jj][1] + Rp[i][2]*ac2*Rp[jj][2];
      }

    auto Aelem = [&](int r, int c) -> float {
      if (r > 8 || c > 8) return 0.f;
      if (c < 3) {
        if (r < 3) return Rdr[c][r];                   // Rdr^T
        if (r < 6) return -RpK[r-3][c] * d;
        return -0.5f * RpK[r-6][c] * d * d;
      }
      if (c < 6) {
        if (r >= 3 && r < 6) return (r - 3 == c - 3) ? 1.f : 0.f;
        if (r >= 6)          return (r - 6 == c - 3) ? d : 0.f;
        return 0.f;
      }
      return (r == c) ? 1.f : 0.f;
    };
    auto Qelem = [&](int r, int c) -> float {
      if (r > 8 || c > 8) return 0.f;
      if (r < 3 && c < 3) return d * Sg[r][c];
      if (r < 3 || c < 3) return 0.f;
      int rb = (r >= 6), cb = (c >= 6);
      float sv = Sa[r - 3 - 3*rb][c - 3 - 3*cb];
      if (!rb && !cb) return d * sv;
      if (rb && cb)   return 0.25f * d * d * d * sv;
      return 0.5f * d * d * sv;
    };

    // A in WMMA A-operand layout; same regs double as B-operand of A^T
    v2f aA[3];
    #pragma unroll
    for (int kk = 0; kk < 3; ++kk) {
      int c0 = 4*kk + 2*hi;
      aA[kk].x = Aelem(m, c0);
      aA[kk].y = Aelem(m, c0 + 1);
    }
    v8f Qr;
    #pragma unroll
    for (int v = 0; v < 8; ++v) Qr[v] = Qelem(v + 8*hi, m);

    // M' = A @ M ; T = A @ S  (K-chunks 0..2; A cols >= 9 are zero)
    v8f Mn = {}; v8f Tt = {};
    #pragma unroll
    for (int kk = 0; kk < 3; ++kk) {
      Mn = wmma4(aA[kk], ldB(tM, kk, hi, m), Mn);
      Tt = wmma4(aA[kk], ldB(tS, kk, hi, m), Tt);
    }
    __syncthreads();
    #pragma unroll
    for (int v = 0; v < 8; ++v) tT[v*32 + lane] = Tt[v];
    __syncthreads();
    // S' = T @ A^T + Q
    v8f Sn = Qr;
    #pragma unroll
    for (int kk = 0; kk < 3; ++kk) Sn = wmma4(ldA(tT, kk, hi, m), aA[kk], Sn);

    M = Mn; S = Sn;
  }

  float* outp = segres + (size_t)(b * kNseg + s) * 512;
  #pragma unroll
  for (int v = 0; v < 8; ++v) {
    outp[v*32 + lane]       = M[v];
    outp[256 + v*32 + lane] = S[v];
  }
}

// ---------------------------------------------------------------------------
// Kernel 3: fold segment results; write final 9x9 covariance
// ---------------------------------------------------------------------------
__global__ __launch_bounds__(32) void imu_fold_kernel(const float* __restrict__ segres,
                                                      float* __restrict__ cov) {
  __shared__ float ldsM[256], ldsS[256], ldsT[256];
  const int b = blockIdx.x;
  const int lane = threadIdx.x;
  const int m = lane & 15;
  const int hi = lane >> 4;

  v8f M, S = {};
  #pragma unroll
  for (int v = 0; v < 8; ++v) M[v] = ((v + 8*hi) == m) ? 1.f : 0.f;

  for (int s = 0; s < kNseg; ++s) {
    const float* Mw = segres + (size_t)(b * kNseg + s) * 512;
    const float* Sw = Mw + 256;

    #pragma unroll
    for (int v = 0; v < 8; ++v) { ldsM[v*32 + lane] = M[v]; ldsS[v*32 + lane] = S[v]; }
    __syncthreads();

    v2f a2[3];
    #pragma unroll
    for (int kk = 0; kk < 3; ++kk) {
      int c0 = 4*kk + 2*hi;
      a2[kk].x = Mw[tidx(m, c0)];
      a2[kk].y = Mw[tidx(m, c0 + 1)];
    }
    v8f S2;
    #pragma unroll
    for (int v = 0; v < 8; ++v) S2[v] = Sw[v*32 + lane];

    v8f Mn = {}; v8f Tt = {};
    #pragma unroll
    for (int kk = 0; kk < 3; ++kk) {
      Mn = wmma4(a2[kk], ldB(ldsM, kk, hi, m), Mn);
      Tt = wmma4(a2[kk], ldB(ldsS, kk, hi, m), Tt);
    }
    __syncthreads();
    #pragma unroll
    for (int v = 0; v < 8; ++v) ldsT[v*32 + lane] = Tt[v];
    __syncthreads();
    v8f Sn = S2;
    #pragma unroll
    for (int kk = 0; kk < 3; ++kk) Sn = wmma4(ldA(ldsT, kk, hi, m), a2[kk], Sn);

    M = Mn; S = Sn;
  }

  #pragma unroll
  for (int v = 0; v < 8; ++v) {
    int r = v + 8*hi, c = m;
    if (r < 9 && c < 9) cov[b*81 + r*9 + c] = S[v];
  }
}

extern "C" void kernel_launch(void* const* d_in, const int* in_sizes, int n_in,
                              void* d_out, int out_size, void* d_ws, size_t ws_size,
                              hipStream_t stream) {
  (void)in_sizes; (void)n_in; (void)out_size; (void)ws_size;
  const float* dt       = (const float*)d_in[0];
  const float* gyro     = (const float*)d_in[1];
  const float* acc      = (const float*)d_in[2];
  const float* init_pos = (const float*)d_in[3];
  const float* init_rot = (const float*)d_in[4];
  const float* init_vel = (const float*)d_in[5];
  const float* gyro_cov = (const float*)d_in[6];
  const float* acc_cov  = (const float*)d_in[7];

  float* out = (float*)d_out;
  float* rot = out;
  float* vel = out + OFF_VEL;
  float* pos = out + OFF_POS;
  float* cov = out + OFF_COV;

  float* qws    = (float*)d_ws;                 // B*F*4 floats: normalized incre_r
  float* segres = qws + (size_t)kB * kF * 4;    // B*NSEG*512 floats: (M,S) per segment

  imu_scan_kernel<<<kB, 256, 0, stream>>>(dt, gyro, acc, init_pos, init_rot, init_vel,
                                          rot, vel, pos, qws);
  imu_seg_kernel<<<kB * kNseg, 32, 0, stream>>>(dt, gyro, acc, gyro_cov, acc_cov, qws, segres);
  imu_fold_kernel<<<kB, 32, 0, stream>>>(segres, cov);
}